// LaplacianUnit_31473520345757
// MI455X (gfx1250) — compile-verified
//
#include <hip/hip_runtime.h>

typedef __attribute__((ext_vector_type(2))) float v2f;
typedef __attribute__((ext_vector_type(8))) float v8f;

#define BB 4
#define NN 16384
#define CC 128
#define KK 16
#define BN_EPS_F 1e-5f
#define KNN_TILE 2048

// ---------------- 1) KNN: per-thread top-16 scan with LDS-staged xyz tiles ----
__global__ void __launch_bounds__(256) knn_kernel(const float* __restrict__ xyz,
                                                  int* __restrict__ idx_out) {
  __shared__ float sx[KNN_TILE * 3];
  const int b = blockIdx.y;
  const int n = blockIdx.x * blockDim.x + threadIdx.x;
  const float* base = xyz + (size_t)b * NN * 3;
  const float qx = base[n * 3 + 0];
  const float qy = base[n * 3 + 1];
  const float qz = base[n * 3 + 2];

  float bd[KK];
  int   bi[KK];
#pragma unroll
  for (int s = 0; s < KK; ++s) { bd[s] = 3.4e38f; bi[s] = 0; }
  float worst = 3.4e38f;
  int   wslot = 0;

  for (int t0 = 0; t0 < NN; t0 += KNN_TILE) {
    __syncthreads();
    for (int i = threadIdx.x; i < KNN_TILE * 3; i += blockDim.x)
      sx[i] = base[t0 * 3 + i];
    __syncthreads();
    for (int jj = 0; jj < KNN_TILE; ++jj) {
      // avoid FMA contraction so distances match the JAX fp32 reference ordering
      float dx = __fsub_rn(qx, sx[jj * 3 + 0]);
      float dy = __fsub_rn(qy, sx[jj * 3 + 1]);
      float dz = __fsub_rn(qz, sx[jj * 3 + 2]);
      float d  = __fadd_rn(__fadd_rn(__fmul_rn(dx, dx), __fmul_rn(dy, dy)),
                           __fmul_rn(dz, dz));
      if (d < worst) {  // strict <: lowest index wins ties, like lax.top_k
#pragma unroll
        for (int s = 0; s < KK; ++s)
          if (s == wslot) { bd[s] = d; bi[s] = t0 + jj; }
        worst = bd[0]; wslot = 0;
#pragma unroll
        for (int s = 1; s < KK; ++s)
          if (bd[s] > worst) { worst = bd[s]; wslot = s; }
      }
    }
  }
  int* o = idx_out + ((size_t)b * NN + n) * KK;
#pragma unroll
  for (int s = 0; s < KK; ++s) o[s] = bi[s];
}

// ---------------- 2) Gather-sum: dx[b,c,n] = sum_j x[b,c,idx[j]] - x[b,c,n] ---
__global__ void __launch_bounds__(256) gather_kernel(const float* __restrict__ x,
                                                     const int* __restrict__ idx,
                                                     float* __restrict__ dx) {
  const int b = blockIdx.y;
  const int n = blockIdx.x * blockDim.x + threadIdx.x;
  int id[KK];
  const int* ip = idx + ((size_t)b * NN + n) * KK;
#pragma unroll
  for (int s = 0; s < KK; ++s) id[s] = ip[s];
  const float* xb  = x  + (size_t)b * CC * NN;
  float*       dxb = dx + (size_t)b * CC * NN;
  for (int c = 0; c < CC; ++c) {
    const float* row = xb + (size_t)c * NN;
    float s = 0.f;
#pragma unroll
    for (int j = 0; j < KK; ++j) s += row[id[j]];
    dxb[(size_t)c * NN + n] = s - row[n];
  }
}

// ---------------- 3) conv1x1 via V_WMMA_F32_16X16X4_F32 + bias + ReLU ---------
// Block: 256 threads (8 waves). Wave w owns output rows o0=16w..16w+15 and the
// block's 64 n-columns (4 x 16x16 tiles). dx tile (128x64) staged in LDS.
// h overwrites dx in place (block reads only its own columns before writing).
__global__ void __launch_bounds__(256) gemm_kernel(float* __restrict__ dxh,
                                                   const float* __restrict__ W,
                                                   const float* __restrict__ bias) {
  __shared__ float lds[CC * 64];
  const int b  = blockIdx.y;
  const int n0 = blockIdx.x * 64;
  const int tid = threadIdx.x;
  float* dxb = dxh + (size_t)b * CC * NN + n0;

  for (int i = tid; i < CC * 16; i += 256) {       // 128 rows x 16 float4
    const int c = i >> 4, q = i & 15;
    const float4 v = *(const float4*)(dxb + (size_t)c * NN + q * 4);
    *(float4*)(lds + c * 64 + q * 4) = v;
  }
  __syncthreads();

  const int lane = tid & 31;
  const int j    = lane & 15;   // column within 16-wide tile / row of A
  const int half = lane >> 4;   // K half-split per ISA f32 A-layout
  const int o0   = (tid >> 5) * 16;

  v8f acc[4] = {};
  const float* wrow = W + (size_t)(o0 + j) * CC + 2 * half;
  for (int k0 = 0; k0 < CC; k0 += 4) {
    v2f a;
    a.x = wrow[k0];
    a.y = wrow[k0 + 1];
    const int krow = k0 + 2 * half;
#pragma unroll
    for (int nt = 0; nt < 4; ++nt) {
      v2f bf;
      bf.x = lds[krow * 64 + nt * 16 + j];
      bf.y = lds[(krow + 1) * 64 + nt * 16 + j];
      acc[nt] = __builtin_amdgcn_wmma_f32_16x16x4_f32(
          /*neg_a=*/false, a, /*neg_b=*/false, bf,
          /*c_mod=*/(short)0, acc[nt], /*reuse_a=*/false, /*reuse_b=*/false);
    }
  }

  float* hb = dxh + (size_t)b * CC * NN + n0;
#pragma unroll
  for (int nt = 0; nt < 4; ++nt) {
#pragma unroll
    for (int r = 0; r < 8; ++r) {
      const int o = o0 + r + 8 * half;             // C/D layout: VGPR r -> M=r / M=8+r
      float v = acc[nt][r] + bias[o];
      v = fmaxf(v, 0.f);
      hb[(size_t)o * NN + nt * 16 + j] = v;
    }
  }
}

// ---------------- 4) BN stats: mean/var per channel over (B,N) ----------------
__global__ void __launch_bounds__(256) bnstats_kernel(const float* __restrict__ h,
                                                      float* __restrict__ stats) {
  __shared__ float s1[256];
  __shared__ float s2[256];
  const int o = blockIdx.x;
  float a = 0.f, q = 0.f;
  for (int i = threadIdx.x; i < BB * NN; i += 256) {
    const int b = i >> 14;          // / NN
    const int n = i & (NN - 1);
    const float v = h[((size_t)b * CC + o) * NN + n];
    a += v;
    q += v * v;
  }
  s1[threadIdx.x] = a;
  s2[threadIdx.x] = q;
  __syncthreads();
  for (int st = 128; st > 0; st >>= 1) {
    if (threadIdx.x < st) {
      s1[threadIdx.x] += s1[threadIdx.x + st];
      s2[threadIdx.x] += s2[threadIdx.x + st];
    }
    __syncthreads();
  }
  if (threadIdx.x == 0) {
    const float inv  = 1.f / (float)(BB * NN);
    const float mean = s1[0] * inv;
    const float var  = s2[0] * inv - mean * mean;
    stats[o]      = mean;
    stats[CC + o] = var;
  }
}

// ---------------- 5) residual + affine BN apply -------------------------------
__global__ void __launch_bounds__(256) finalize_kernel(const float* __restrict__ x,
                                                       const float* __restrict__ h,
                                                       const float* __restrict__ stats,
                                                       const float* __restrict__ gamma,
                                                       const float* __restrict__ beta,
                                                       float* __restrict__ out) {
  const size_t i = (size_t)blockIdx.x * blockDim.x + threadIdx.x;
  if (i >= (size_t)BB * CC * NN) return;
  const int c = (int)((i >> 14) & (CC - 1));   // (i / N) % C
  const float mean = stats[c];
  const float var  = stats[CC + c];
  const float hn   = (h[i] - mean) * rsqrtf(var + BN_EPS_F);
  out[i] = x[i] + gamma[c] * hn + beta[c];
}

extern "C" void kernel_launch(void* const* d_in, const int* in_sizes, int n_in,
                              void* d_out, int out_size, void* d_ws, size_t ws_size,
                              hipStream_t stream) {
  const float* xyz   = (const float*)d_in[0];
  const float* x     = (const float*)d_in[1];
  const float* W     = (const float*)d_in[2];
  const float* bias  = (const float*)d_in[3];
  const float* gamma = (const float*)d_in[4];
  const float* beta  = (const float*)d_in[5];
  float* out = (float*)d_out;

  // workspace: idx (4MB) | dx-aliased-h (32MB) | stats (1KB)  => ~36MB
  char* ws = (char*)d_ws;
  int*   idx_ws = (int*)ws;
  float* dxh_ws = (float*)(ws + (size_t)BB * NN * KK * sizeof(int));
  float* stats  = (float*)(ws + (size_t)BB * NN * KK * sizeof(int)
                              + (size_t)BB * CC * NN * sizeof(float));

  knn_kernel   <<<dim3(NN / 256, BB), 256, 0, stream>>>(xyz, idx_ws);
  gather_kernel<<<dim3(NN / 256, BB), 256, 0, stream>>>(x, idx_ws, dxh_ws);
  gemm_kernel  <<<dim3(NN / 64,  BB), 256, 0, stream>>>(dxh_ws, W, bias);
  bnstats_kernel<<<CC, 256, 0, stream>>>(dxh_ws, stats);
  const int total = BB * CC * NN;
  finalize_kernel<<<(total + 255) / 256, 256, 0, stream>>>(x, dxh_ws, stats,
                                                           gamma, beta, out);
}